// GAT_NET_57913339019857
// MI455X (gfx1250) — compile-verified
//
#include <hip/hip_runtime.h>

typedef float v2f __attribute__((ext_vector_type(2)));
typedef float v8f __attribute__((ext_vector_type(8)));

#define FIN   128
#define C1    256   // HEADS*HID
#define C2    64    // CLASSES
#define SLOPE_ATT 0.2f
#define SLOPE_ACT 0.01f
#define BN_EPS 1e-5f

// ---------- helpers ----------
__device__ __forceinline__ unsigned fkey(float f) {
  unsigned u = __float_as_uint(f);
  return (u & 0x80000000u) ? ~u : (u | 0x80000000u);
}
__device__ __forceinline__ float funkey(unsigned u) {
  return (u & 0x80000000u) ? __uint_as_float(u & 0x7fffffffu) : __uint_as_float(~u);
}
__device__ __forceinline__ float lrelu(float x, float s) { return x >= 0.f ? x : s * x; }

// ---------- fp32 WMMA GEMM: D[M,ncols] = A[M,K] @ B[K,ncols] ----------
// One wave computes a 64(M) x 16(N) tile: 4 WMMA accumulators sharing each B load.
// grid.x = ceil(M/64); blockDim.x = (ncols/16)*32 ; wave w owns N-tile w.
__global__ void wmma_gemm_f32(const float* __restrict__ A, const float* __restrict__ B,
                              float* __restrict__ D, int M, int K, int ncols) {
  int wave = threadIdx.x >> 5;
  int lane = threadIdx.x & 31;
  int half = lane >> 4;
  int idx  = lane & 15;
  int m0 = blockIdx.x << 6;
  int n0 = wave << 4;

  const float* Ap0; const float* Ap1; const float* Ap2; const float* Ap3;
  {
    int r0 = m0 +  0 + idx; if (r0 >= M) r0 = M - 1;
    int r1 = m0 + 16 + idx; if (r1 >= M) r1 = M - 1;
    int r2 = m0 + 32 + idx; if (r2 >= M) r2 = M - 1;
    int r3 = m0 + 48 + idx; if (r3 >= M) r3 = M - 1;
    Ap0 = A + (size_t)r0 * K;  Ap1 = A + (size_t)r1 * K;
    Ap2 = A + (size_t)r2 * K;  Ap3 = A + (size_t)r3 * K;
  }
  const float* Bp = B + n0 + idx;

  v8f acc0 = {0.f,0.f,0.f,0.f,0.f,0.f,0.f,0.f};
  v8f acc1 = acc0, acc2 = acc0, acc3 = acc0;

  for (int k = 0; k < K; k += 4) {
    int kl = k + 2 * half;
    v2f b;
    b.x = Bp[(size_t)kl * ncols];                   // B VGPR j -> K = kl+j
    b.y = Bp[(size_t)(kl + 1) * ncols];
    v2f a0 = *(const v2f*)(Ap0 + kl);               // A: row in lane, K = {kl, kl+1}
    v2f a1 = *(const v2f*)(Ap1 + kl);
    v2f a2 = *(const v2f*)(Ap2 + kl);
    v2f a3 = *(const v2f*)(Ap3 + kl);
    if (k + 4 < K) {                                // prefetch next B rows
      __builtin_prefetch(Bp + (size_t)(kl + 4) * ncols, 0, 3);
      __builtin_prefetch(Bp + (size_t)(kl + 5) * ncols, 0, 3);
    }
    acc0 = __builtin_amdgcn_wmma_f32_16x16x4_f32(false, a0, false, b, (short)0, acc0, false, false);
    acc1 = __builtin_amdgcn_wmma_f32_16x16x4_f32(false, a1, false, b, (short)0, acc1, false, false);
    acc2 = __builtin_amdgcn_wmma_f32_16x16x4_f32(false, a2, false, b, (short)0, acc2, false, false);
    acc3 = __builtin_amdgcn_wmma_f32_16x16x4_f32(false, a3, false, b, (short)0, acc3, false, false);
  }

  float* Dp = D + n0 + idx;
  #pragma unroll
  for (int v = 0; v < 8; ++v) {
    int rr = v + 8 * half;
    int r0 = m0 + rr, r1 = m0 + 16 + rr, r2 = m0 + 32 + rr, r3 = m0 + 48 + rr;
    if (r0 < M) Dp[(size_t)r0 * ncols] = acc0[v];
    if (r1 < M) Dp[(size_t)r1 * ncols] = acc1[v];
    if (r2 < M) Dp[(size_t)r2 * ncols] = acc2[v];
    if (r3 < M) Dp[(size_t)r3 * ncols] = acc3[v];
  }
}

// ---------- per-node attention coefficients, layer 1 (4 heads x 64) ----------
__global__ void node_alpha4(const float* __restrict__ H, const float* __restrict__ asr,
                            const float* __restrict__ ads, float* __restrict__ as_,
                            float* __restrict__ ad_, int Nn) {
  int wave = threadIdx.x >> 5, lane = threadIdx.x & 31;
  int n = blockIdx.x * (blockDim.x >> 5) + wave;
  if (n >= Nn) return;
  const float* hr = H + (size_t)n * C1;
  float ss[4] = {0,0,0,0}, sd[4] = {0,0,0,0};
  #pragma unroll
  for (int i = 0; i < 8; ++i) {
    int ch = lane + 32 * i;           // head = i>>1
    float v = hr[ch];
    ss[i >> 1] += v * asr[ch];
    sd[i >> 1] += v * ads[ch];
  }
  #pragma unroll
  for (int h = 0; h < 4; ++h) {
    float s = ss[h], d = sd[h];
    for (int off = 16; off > 0; off >>= 1) {
      s += __shfl_down(s, off, 32);
      d += __shfl_down(d, off, 32);
    }
    if (lane == 0) { as_[(size_t)n * 4 + h] = s; ad_[(size_t)n * 4 + h] = d; }
  }
}

// ---------- per-node attention coefficients, layer 2 (1 head x 64) ----------
__global__ void node_alpha1(const float* __restrict__ H, const float* __restrict__ asr,
                            const float* __restrict__ ads, float* __restrict__ as_,
                            float* __restrict__ ad_, int Nn) {
  int wave = threadIdx.x >> 5, lane = threadIdx.x & 31;
  int n = blockIdx.x * (blockDim.x >> 5) + wave;
  if (n >= Nn) return;
  const float* hr = H + (size_t)n * C2;
  float s = hr[lane] * asr[lane] + hr[lane + 32] * asr[lane + 32];
  float d = hr[lane] * ads[lane] + hr[lane + 32] * ads[lane + 32];
  for (int off = 16; off > 0; off >>= 1) {
    s += __shfl_down(s, off, 32);
    d += __shfl_down(d, off, 32);
  }
  if (lane == 0) { as_[n] = s; ad_[n] = d; }
}

// ---------- edge helpers ----------
__device__ __forceinline__ void edge_sd(int e, int E_, const int* __restrict__ src,
                                        const int* __restrict__ dst, int& s, int& d) {
  if (e < E_) { s = src[e]; d = dst[e]; } else { s = e - E_; d = s; }
}

// ---------- layer-1 edge passes (4 heads) ----------
__global__ void edge_max4(const int* __restrict__ src, const int* __restrict__ dst,
                          const float* __restrict__ as_, const float* __restrict__ ad_,
                          unsigned* __restrict__ mkey, int E_, int Et) {
  int e = blockIdx.x * blockDim.x + threadIdx.x;
  if (e >= Et) return;
  int s, d; edge_sd(e, E_, src, dst, s, d);
  float4 A = *(const float4*)(as_ + (size_t)s * 4);
  float4 B = *(const float4*)(ad_ + (size_t)d * 4);
  unsigned* mk = mkey + (size_t)d * 4;
  atomicMax(mk + 0, fkey(lrelu(A.x + B.x, SLOPE_ATT)));
  atomicMax(mk + 1, fkey(lrelu(A.y + B.y, SLOPE_ATT)));
  atomicMax(mk + 2, fkey(lrelu(A.z + B.z, SLOPE_ATT)));
  atomicMax(mk + 3, fkey(lrelu(A.w + B.w, SLOPE_ATT)));
}

__global__ void edge_sum4(const int* __restrict__ src, const int* __restrict__ dst,
                          const float* __restrict__ as_, const float* __restrict__ ad_,
                          const unsigned* __restrict__ mkey, float* __restrict__ den,
                          int E_, int Et) {
  int e = blockIdx.x * blockDim.x + threadIdx.x;
  if (e >= Et) return;
  int s, d; edge_sd(e, E_, src, dst, s, d);
  float4 A = *(const float4*)(as_ + (size_t)s * 4);
  float4 B = *(const float4*)(ad_ + (size_t)d * 4);
  const unsigned* mk = mkey + (size_t)d * 4;
  float* dn = den + (size_t)d * 4;
  atomicAdd(dn + 0, __expf(lrelu(A.x + B.x, SLOPE_ATT) - funkey(mk[0])));
  atomicAdd(dn + 1, __expf(lrelu(A.y + B.y, SLOPE_ATT) - funkey(mk[1])));
  atomicAdd(dn + 2, __expf(lrelu(A.z + B.z, SLOPE_ATT) - funkey(mk[2])));
  atomicAdd(dn + 3, __expf(lrelu(A.w + B.w, SLOPE_ATT) - funkey(mk[3])));
}

__global__ void edge_att4(const int* __restrict__ src, const int* __restrict__ dst,
                          const float* __restrict__ as_, const float* __restrict__ ad_,
                          const unsigned* __restrict__ mkey, const float* __restrict__ den,
                          float* __restrict__ att, int E_, int Et) {
  int e = blockIdx.x * blockDim.x + threadIdx.x;
  if (e >= Et) return;
  int s, d; edge_sd(e, E_, src, dst, s, d);
  float4 A = *(const float4*)(as_ + (size_t)s * 4);
  float4 B = *(const float4*)(ad_ + (size_t)d * 4);
  const unsigned* mk = mkey + (size_t)d * 4;
  const float* dn = den + (size_t)d * 4;
  float4 o;
  o.x = __expf(lrelu(A.x + B.x, SLOPE_ATT) - funkey(mk[0])) / dn[0];
  o.y = __expf(lrelu(A.y + B.y, SLOPE_ATT) - funkey(mk[1])) / dn[1];
  o.z = __expf(lrelu(A.z + B.z, SLOPE_ATT) - funkey(mk[2])) / dn[2];
  o.w = __expf(lrelu(A.w + B.w, SLOPE_ATT) - funkey(mk[3])) / dn[3];
  *(float4*)(att + (size_t)e * 4) = o;
}

// 64 threads/edge; thread c handles channels {c, c+64, c+128, c+192} (one per head)
__global__ void edge_scatter4(const int* __restrict__ src, const int* __restrict__ dst,
                              const float* __restrict__ H, const float* __restrict__ att,
                              float* __restrict__ out, int E_, int Et) {
  int gid = blockIdx.x * blockDim.x + threadIdx.x;
  int e = gid >> 6, c = gid & 63;
  if (e >= Et) return;
  int s, d; edge_sd(e, E_, src, dst, s, d);
  float4 a = *(const float4*)(att + (size_t)e * 4);
  const float* hs = H + (size_t)s * C1 + c;
  float* od = out + (size_t)d * C1 + c;
  atomicAdd(od +   0, hs[  0] * a.x);
  atomicAdd(od +  64, hs[ 64] * a.y);
  atomicAdd(od + 128, hs[128] * a.z);
  atomicAdd(od + 192, hs[192] * a.w);
}

// ---------- layer-2 edge passes (1 head) ----------
__global__ void edge_max1(const int* __restrict__ src, const int* __restrict__ dst,
                          const float* __restrict__ as_, const float* __restrict__ ad_,
                          unsigned* __restrict__ mkey, int E_, int Et) {
  int e = blockIdx.x * blockDim.x + threadIdx.x;
  if (e >= Et) return;
  int s, d; edge_sd(e, E_, src, dst, s, d);
  atomicMax(mkey + d, fkey(lrelu(as_[s] + ad_[d], SLOPE_ATT)));
}

__global__ void edge_sum1(const int* __restrict__ src, const int* __restrict__ dst,
                          const float* __restrict__ as_, const float* __restrict__ ad_,
                          const unsigned* __restrict__ mkey, float* __restrict__ den,
                          int E_, int Et) {
  int e = blockIdx.x * blockDim.x + threadIdx.x;
  if (e >= Et) return;
  int s, d; edge_sd(e, E_, src, dst, s, d);
  atomicAdd(den + d, __expf(lrelu(as_[s] + ad_[d], SLOPE_ATT) - funkey(mkey[d])));
}

__global__ void edge_att1(const int* __restrict__ src, const int* __restrict__ dst,
                          const float* __restrict__ as_, const float* __restrict__ ad_,
                          const unsigned* __restrict__ mkey, const float* __restrict__ den,
                          float* __restrict__ att, int E_, int Et) {
  int e = blockIdx.x * blockDim.x + threadIdx.x;
  if (e >= Et) return;
  int s, d; edge_sd(e, E_, src, dst, s, d);
  att[e] = __expf(lrelu(as_[s] + ad_[d], SLOPE_ATT) - funkey(mkey[d])) / den[d];
}

__global__ void edge_scatter1(const int* __restrict__ src, const int* __restrict__ dst,
                              const float* __restrict__ H, const float* __restrict__ att,
                              float* __restrict__ out, int E_, int Et) {
  int gid = blockIdx.x * blockDim.x + threadIdx.x;
  int e = gid >> 6, c = gid & 63;
  if (e >= Et) return;
  int s, d; edge_sd(e, E_, src, dst, s, d);
  atomicAdd(out + (size_t)d * C2 + c, H[(size_t)s * C2 + c] * att[e]);
}

// ---------- BatchNorm ----------
__global__ void bn_stats(const float* __restrict__ H, float* __restrict__ sum,
                         float* __restrict__ sq, int Nn) {
  int c = threadIdx.x;                       // 256 threads = one per channel
  int r0 = blockIdx.x * 64;
  int r1 = r0 + 64; if (r1 > Nn) r1 = Nn;
  float s = 0.f, s2 = 0.f;
  for (int r = r0; r < r1; ++r) {
    float v = H[(size_t)r * C1 + c];
    s += v; s2 += v * v;
  }
  atomicAdd(sum + c, s);
  atomicAdd(sq + c, s2);
}

__global__ void bn_finalize(const float* __restrict__ sum, const float* __restrict__ sq,
                            const float* __restrict__ gamma, const float* __restrict__ beta,
                            float* __restrict__ scale, float* __restrict__ shift, float invN) {
  int c = threadIdx.x;
  float mean = sum[c] * invN;
  float var  = sq[c] * invN - mean * mean;
  float inv  = rsqrtf(var + BN_EPS);
  float sc   = gamma[c] * inv;
  scale[c] = sc;
  shift[c] = beta[c] - mean * sc;
}

// float4-vectorized: total4 = N*C1/4 elements
__global__ void bn_apply(float* __restrict__ H, const float* __restrict__ scale,
                         const float* __restrict__ shift, int total4) {
  int i = blockIdx.x * blockDim.x + threadIdx.x;
  if (i >= total4) return;
  int c = (i * 4) & (C1 - 1);
  float4 v = ((const float4*)H)[i];
  float4 sc = *(const float4*)(scale + c);
  float4 sh = *(const float4*)(shift + c);
  v.x = lrelu(v.x * sc.x + sh.x, SLOPE_ACT);
  v.y = lrelu(v.y * sc.y + sh.y, SLOPE_ACT);
  v.z = lrelu(v.z * sc.z + sh.z, SLOPE_ACT);
  v.w = lrelu(v.w * sc.w + sh.w, SLOPE_ACT);
  ((float4*)H)[i] = v;
}

// float4-vectorized bias add over [N, C2]
__global__ void bias_add(float* __restrict__ out, const float* __restrict__ b, int total4) {
  int i = blockIdx.x * blockDim.x + threadIdx.x;
  if (i >= total4) return;
  int c = (i * 4) & (C2 - 1);
  float4 v = ((const float4*)out)[i];
  float4 bb = *(const float4*)(b + c);
  v.x += bb.x; v.y += bb.y; v.z += bb.z; v.w += bb.w;
  ((float4*)out)[i] = v;
}

// ---------- host ----------
extern "C" void kernel_launch(void* const* d_in, const int* in_sizes, int n_in,
                              void* d_out, int out_size, void* d_ws, size_t ws_size,
                              hipStream_t stream) {
  const float* x      = (const float*)d_in[0];
  const int*   ei     = (const int*)  d_in[1];
  const float* W1     = (const float*)d_in[2];
  const float* a_src1 = (const float*)d_in[3];
  const float* a_dst1 = (const float*)d_in[4];
  // d_in[5] = b1: cancels exactly through BatchNorm (shift-invariant) -> skipped
  const float* gamma  = (const float*)d_in[6];
  const float* beta   = (const float*)d_in[7];
  const float* W2     = (const float*)d_in[8];
  const float* a_src2 = (const float*)d_in[9];
  const float* a_dst2 = (const float*)d_in[10];
  const float* b2     = (const float*)d_in[11];

  const int Nn = in_sizes[0] / FIN;
  const int Ee = in_sizes[1] / 2;
  const int Et = Ee + Nn;                   // edges + self-loops
  const int* srcI = ei;
  const int* dstI = ei + Ee;

  char* w = (char*)d_ws;
  size_t off = 0;
  auto take = [&](size_t bytes) -> void* {
    void* p = w + off;
    off = (off + bytes + 255) & ~(size_t)255;
    return p;
  };
  float*    h1    = (float*)   take((size_t)Nn * C1 * 4);
  float*    out1  = (float*)   take((size_t)Nn * C1 * 4);
  float*    as1   = (float*)   take((size_t)Nn * 4 * 4);
  float*    ad1   = (float*)   take((size_t)Nn * 4 * 4);
  unsigned* mk1   = (unsigned*)take((size_t)Nn * 4 * 4);
  float*    den1  = (float*)   take((size_t)Nn * 4 * 4);
  float*    att1b = (float*)   take((size_t)Et * 4 * 4);
  float*    bnsum = (float*)   take(C1 * 4);
  float*    bnsq  = (float*)   take(C1 * 4);
  float*    scale = (float*)   take(C1 * 4);
  float*    shift = (float*)   take(C1 * 4);
  float*    h2    = (float*)   take((size_t)Nn * C2 * 4);
  float*    as2   = (float*)   take((size_t)Nn * 4);
  float*    ad2   = (float*)   take((size_t)Nn * 4);
  unsigned* mk2   = (unsigned*)take((size_t)Nn * 4);
  float*    den2  = (float*)   take((size_t)Nn * 4);
  float*    att2b = (float*)   take((size_t)Et * 4);
  float*    out2  = (float*)d_out;

  // zero all accumulators (graph-capture-legal async memsets)
  hipMemsetAsync(out1,  0, (size_t)Nn * C1 * 4, stream);
  hipMemsetAsync(mk1,   0, (size_t)Nn * 4 * 4,  stream);
  hipMemsetAsync(den1,  0, (size_t)Nn * 4 * 4,  stream);
  hipMemsetAsync(bnsum, 0, C1 * 4,              stream);
  hipMemsetAsync(bnsq,  0, C1 * 4,              stream);
  hipMemsetAsync(mk2,   0, (size_t)Nn * 4,      stream);
  hipMemsetAsync(den2,  0, (size_t)Nn * 4,      stream);
  hipMemsetAsync(out2,  0, (size_t)out_size * 4, stream);

  const int mtiles64 = (Nn + 63) / 64;
  const int egrid    = (Et + 255) / 256;
  const int sgrid    = ((size_t)Et * 64 + 255) / 256;

  // ===== layer 1 =====
  wmma_gemm_f32<<<mtiles64, (C1 / 16) * 32, 0, stream>>>(x, W1, h1, Nn, FIN, C1);
  node_alpha4<<<(Nn + 7) / 8, 256, 0, stream>>>(h1, a_src1, a_dst1, as1, ad1, Nn);
  edge_max4<<<egrid, 256, 0, stream>>>(srcI, dstI, as1, ad1, mk1, Ee, Et);
  edge_sum4<<<egrid, 256, 0, stream>>>(srcI, dstI, as1, ad1, mk1, den1, Ee, Et);
  edge_att4<<<egrid, 256, 0, stream>>>(srcI, dstI, as1, ad1, mk1, den1, att1b, Ee, Et);
  edge_scatter4<<<sgrid, 256, 0, stream>>>(srcI, dstI, h1, att1b, out1, Ee, Et);

  // ===== BatchNorm (train mode, batch stats) + leaky_relu, in place on out1 =====
  bn_stats<<<(Nn + 63) / 64, C1, 0, stream>>>(out1, bnsum, bnsq, Nn);
  bn_finalize<<<1, C1, 0, stream>>>(bnsum, bnsq, gamma, beta, scale, shift, 1.0f / (float)Nn);
  {
    int total4 = (Nn * C1) / 4;
    bn_apply<<<(total4 + 255) / 256, 256, 0, stream>>>(out1, scale, shift, total4);
  }

  // ===== layer 2 =====
  wmma_gemm_f32<<<mtiles64, (C2 / 16) * 32, 0, stream>>>(out1, W2, h2, Nn, C1, C2);
  node_alpha1<<<(Nn + 7) / 8, 256, 0, stream>>>(h2, a_src2, a_dst2, as2, ad2, Nn);
  edge_max1<<<egrid, 256, 0, stream>>>(srcI, dstI, as2, ad2, mk2, Ee, Et);
  edge_sum1<<<egrid, 256, 0, stream>>>(srcI, dstI, as2, ad2, mk2, den2, Ee, Et);
  edge_att1<<<egrid, 256, 0, stream>>>(srcI, dstI, as2, ad2, mk2, den2, att2b, Ee, Et);
  edge_scatter1<<<sgrid, 256, 0, stream>>>(srcI, dstI, h2, att2b, out2, Ee, Et);
  {
    int total4 = (Nn * C2) / 4;
    bias_add<<<(total4 + 255) / 256, 256, 0, stream>>>(out2, b2, total4);
  }
}